// ColorConsistencyLoss_90804198572268
// MI455X (gfx1250) — compile-verified
//
#include <hip/hip_runtime.h>
#include <hip/hip_bf16.h>

typedef float f4  __attribute__((ext_vector_type(4)));
typedef float v2f __attribute__((ext_vector_type(2)));
typedef float v8f __attribute__((ext_vector_type(8)));

#define HW      262144   // 512*512, per-channel plane size
#define LOG2HW  18
#define T0f     0.008856f
#define QPT     4        // quads per thread in kernel 1

// cbrt via native v_log_f32 / v_exp_f32 (base-2): t is pre-clamped >= T0 > 0
__device__ __forceinline__ float cbrt_fast(float t) {
    return __builtin_amdgcn_exp2f(0.33333334f * __builtin_amdgcn_logf(t));
}

// Lab pieces for one pixel: scaled L (L/100) and the f() values fx,fy,fz.
// XYZ coefficients are pre-divided by XN / ZN.
__device__ __forceinline__ void lab_parts(float r, float g, float b,
                                          float& Ls, float& fx, float& fy, float& fz) {
    const float x = fmaf(r, 0.43395293f, fmaf(g, 0.37622000f, b * 0.18982780f));
    const float y = fmaf(r, 0.212671f,  fmaf(g, 0.71516f,    b * 0.072169f));
    const float z = fmaf(r, 0.01775806f, fmaf(g, 0.10947670f, b * 0.87276512f));
    const float cx = cbrt_fast(fmaxf(x, T0f));
    const float cy = cbrt_fast(fmaxf(y, T0f));
    const float cz = cbrt_fast(fmaxf(z, T0f));
    fx = (x > T0f) ? cx : fmaf(x, 7.787f, 0.13793103f);
    fy = (y > T0f) ? cy : fmaf(y, 7.787f, 0.13793103f);
    fz = (z > T0f) ? cz : fmaf(z, 7.787f, 0.13793103f);
    Ls = ((y > T0f) ? fmaf(fy, 116.0f, -16.0f) : 903.3f * y) * 0.01f;
}

// Squared Lab-space distance for one pixel pair (output scaling: L/100,
// (a+128)/255, (b+128)/255 -> constant offsets cancel in the difference).
__device__ __forceinline__ float pix_sq(float rp, float gp, float bp,
                                        float rt, float gt, float bt) {
    float Lp, fxp, fyp, fzp, Lt, fxt, fyt, fzt;
    lab_parts(rp, gp, bp, Lp, fxp, fyp, fzp);
    lab_parts(rt, gt, bt, Lt, fxt, fyt, fzt);
    const float dfx = fxp - fxt;
    const float dfy = fyp - fyt;
    const float dfz = fzp - fzt;
    const float dL = Lp - Lt;
    const float da = 1.96078431f * (dfx - dfy);   // 500/255
    const float db = 0.78431373f * (dfy - dfz);   // 200/255
    return fmaf(dL, dL, fmaf(da, da, db * db));
}

// Kernel 1: streaming Lab-MSE partial sums. Each thread handles QPT quads of
// 4 consecutive pixels (6 x 16B non-temporal loads per quad, 24 loads in
// flight), then wave32 shuffle reduce + LDS cross-wave reduce -> one partial
// per block.
__global__ __launch_bounds__(256)
void lab_mse_partial(const float* __restrict__ pred, const float* __restrict__ targ,
                     float* __restrict__ partial, int nquads) {
    const int q0 = blockIdx.x * (256 * QPT) + threadIdx.x;
    float acc = 0.0f;

    #pragma unroll
    for (int k = 0; k < QPT; ++k) {
        const int q = q0 + k * 256;
        if (q < nquads) {
            const int p    = q << 2;             // pixel index, quad-aligned; never
            const int bat  = p >> LOG2HW;        // straddles an image (HW % 4 == 0)
            const int i    = p & (HW - 1);
            const int base = bat * (3 * HW) + i;

            const f4 pr = __builtin_nontemporal_load((const f4*)(pred + base));
            const f4 pg = __builtin_nontemporal_load((const f4*)(pred + base + HW));
            const f4 pb = __builtin_nontemporal_load((const f4*)(pred + base + 2 * HW));
            const f4 tr = __builtin_nontemporal_load((const f4*)(targ + base));
            const f4 tg = __builtin_nontemporal_load((const f4*)(targ + base + HW));
            const f4 tb = __builtin_nontemporal_load((const f4*)(targ + base + 2 * HW));

            #pragma unroll
            for (int j = 0; j < 4; ++j)
                acc += pix_sq(pr[j], pg[j], pb[j], tr[j], tg[j], tb[j]);
        }
    }

    // wave32 reduce
    #pragma unroll
    for (int m = 16; m > 0; m >>= 1) acc += __shfl_xor(acc, m, 32);

    __shared__ float wsum[8];
    if ((threadIdx.x & 31) == 0) wsum[threadIdx.x >> 5] = acc;
    __syncthreads();
    if (threadIdx.x == 0) {
        float s = 0.0f;
        #pragma unroll
        for (int w = 0; w < 8; ++w) s += wsum[w];
        partial[blockIdx.x] = s;
    }
}

// Kernel 2: single block folds the (<=1024) per-block partials. Each thread
// makes ONE independent b128 load (no serialized load chain). The first
// cross-lane fold uses V_WMMA_F32_16X16X4_F32 with A = ones:
// D[0][n] = acc[n] + acc[n+16].
__global__ __launch_bounds__(256)
void lab_mse_final(const float* __restrict__ partial, float* __restrict__ out,
                   int n, float inv_count) {
    const int base = threadIdx.x * 4;
    float s = 0.0f;
    if (base + 3 < n) {
        const f4 p = *(const f4*)(partial + base);
        s = (p[0] + p[1]) + (p[2] + p[3]);
    } else {
        for (int i = base; i < n; ++i) s += partial[i];
    }

    v2f A; A[0] = 1.0f; A[1] = 1.0f;     // 16x4 all-ones
    v2f B; B[0] = s;    B[1] = 0.0f;     // 4x16: rows {acc, 0}
    v8f C = {};
    // (neg_a, A, neg_b, B, c_mod, C, reuse_a, reuse_b)
    v8f D = __builtin_amdgcn_wmma_f32_16x16x4_f32(
        false, A, false, B, (short)0, C, false, false);
    float t = D[0];                       // every lane: s[lane&15] + s[(lane&15)+16]
    t += __shfl_xor(t, 1, 32);
    t += __shfl_xor(t, 2, 32);
    t += __shfl_xor(t, 4, 32);
    t += __shfl_xor(t, 8, 32);            // wave total in all lanes

    __shared__ float wsum[8];
    if ((threadIdx.x & 31) == 0) wsum[threadIdx.x >> 5] = t;
    __syncthreads();
    if (threadIdx.x == 0) {
        float tot = 0.0f;
        #pragma unroll
        for (int w = 0; w < 8; ++w) tot += wsum[w];
        out[0] = tot * inv_count;         // WEIGHT == 1.0
    }
}

extern "C" void kernel_launch(void* const* d_in, const int* in_sizes, int n_in,
                              void* d_out, int out_size, void* d_ws, size_t ws_size,
                              hipStream_t stream) {
    const float* pred = (const float*)d_in[0];
    const float* targ = (const float*)d_in[1];
    float* out = (float*)d_out;
    float* ws  = (float*)d_ws;

    const int total  = in_sizes[0];       // 16*3*512*512 = 12,582,912
    const int pixels = total / 3;         // 4,194,304
    const int nquads = pixels / 4;        // 1,048,576
    const int threads = 256;
    const int qpb = threads * QPT;        // quads per block = 1024
    const int nblocks = (nquads + qpb - 1) / qpb;  // 1024 -> 1024 partials

    lab_mse_partial<<<nblocks, threads, 0, stream>>>(pred, targ, ws, nquads);
    lab_mse_final<<<1, threads, 0, stream>>>(ws, out, nblocks, 1.0f / (float)total);
}